// PointCloudEncoder_19009525252691
// MI455X (gfx1250) — compile-verified
//
#include <hip/hip_runtime.h>

// ---------------------------------------------------------------------------
// PointPillars PFN on gfx1250 (MI455X), wave32 + V_WMMA_F32_16X16X4_F32.
//
// Algebra: A_aug(Px9) = F(Px4)·E + 1_real·c^T  (E fixed 0/1, c per-pillar)
//   =>  D = F·(E·B) + indicator·off^T,  off[n] = -(mx,my,mz,cx,cy)·B[4:9,n]
// K = 5 padded to 8: 7 Mtiles x 4 Ntiles x 2 ksteps = 56 WMMA/pillar.
// ReLU/bias hoisted out of the point-max (monotone): epilogue = pure v_max3.
//
// ws layout (floats): [0,512)   per-channel pack[n][8] = {bias, o4..o8, 0, 0}
//                     [512,768) EB frags: [nt(4)][lane(32)][j(2)]
// ---------------------------------------------------------------------------

typedef __attribute__((ext_vector_type(2))) float v2f;
typedef __attribute__((ext_vector_type(8))) float v8f;

#define NVOX   16000
#define PNTS   100
#define MPAD   112
#define COUT   64
#define CAUG   9
#define BN_EPS 1e-3f
#define VXF    0.16f
#define XOFF   0.08f              // VX/2 + 0.0
#define YOFF   (0.08f - 40.0f)    // VY/2 + (-40)
#define NEG_INF (-3.402823466e38f)

// -------------------- prelude: fold BN + augmentation into W ----------------
__global__ void pfn_prep(const float* __restrict__ W, const float* __restrict__ gamma,
                         const float* __restrict__ beta, const float* __restrict__ rmean,
                         const float* __restrict__ rvar, float* __restrict__ ws) {
  int tid = threadIdx.x;                           // 256 threads
  // packed per-channel constants: 64 channels x 8 floats
  for (int e = tid; e < 512; e += 256) {
    int n = e >> 3, c = e & 7;
    float s = gamma[n] * rsqrtf(rvar[n] + BN_EPS);
    float v = 0.f;
    if (c == 0)      v = beta[n] - rmean[n] * s;   // bias_eff
    else if (c <= 5) v = W[n * CAUG + 3 + c] * s;  // scaled W rows 4..8 (off weights)
    ws[e] = v;
  }
  // EB fragments (K=4 step), B layout v0={K0|K2}, v1={K1|K3}
  if (tid < 256) {
    int e  = tid;
    int nt = e >> 6;
    int l  = (e >> 1) & 31;
    int j  = e & 1;
    int k  = ((l >= 16) ? 2 : 0) + j;              // 0..3
    int n  = nt * 16 + (l & 15);
    float s = gamma[n] * rsqrtf(rvar[n] + BN_EPS);
    const float* w = W + n * CAUG;
    float vch;
    if      (k == 0) vch = w[0] + w[4] + w[7];     // x feeds raw, cluster-x, center-x
    else if (k == 1) vch = w[1] + w[5] + w[8];     // y feeds raw, cluster-y, center-y
    else if (k == 2) vch = w[2] + w[6];            // z feeds raw, cluster-z
    else             vch = w[3];                   // reflectance
    ws[512 + e] = vch * s;
  }
}

// ------------------------------- main kernel --------------------------------
__global__ __launch_bounds__(256) void pfn_main(const float* __restrict__ feats,
                                                const int* __restrict__ indices,
                                                const int* __restrict__ npts_arr,
                                                const float* __restrict__ ws,
                                                float* __restrict__ out) {
  __shared__ alignas(16) float lds[8][MPAD * 4];   // pre-masked feats per wave
  const int lane = threadIdx.x & 31;
  const int wv   = threadIdx.x >> 5;
  const int v    = blockIdx.x * 8 + wv;
  if (v >= NVOX) return;                           // never taken with exact grid

  const int   npts = npts_arr[v];
  const float cx   = (float)indices[v * 3 + 2] * VXF + XOFF;  // indices are ZYX
  const float cy   = (float)indices[v * 3 + 1] * VXF + YOFF;
  const bool  hi   = (lane >= 16);
  const int   nn   = lane & 15;

  // --- per-lane constants: 2x b128 per N-tile + EB fragment (cache-hot) -----
  v2f   Bf0[4];
  float biasn[4], o4[4], o5[4], o6[4], o7[4], o8[4];
#pragma unroll
  for (int nt = 0; nt < 4; ++nt) {
    const float4* pk = (const float4*)(ws + (nt * 16 + nn) * 8);
    float4 p0 = pk[0], p1 = pk[1];
    biasn[nt] = p0.x;
    o4[nt] = p0.y;  o5[nt] = p0.z;  o6[nt] = p0.w;
    o7[nt] = p1.x;  o8[nt] = p1.y;
    Bf0[nt] = *(const v2f*)(ws + 512 + (nt * 32 + lane) * 2);
  }

  // --- stage masked features to LDS (zeros for padded + pad rows), sums ------
  float sx = 0.f, sy = 0.f, sz = 0.f;
  const float* fv = feats + (size_t)v * (PNTS * 4);
#pragma unroll
  for (int i = 0; i < 4; ++i) {
    int p = lane + 32 * i;
    if (p < MPAD) {
      float4 f = make_float4(0.f, 0.f, 0.f, 0.f);
      if (p < npts) {                              // mask applied at fill
        f = *(const float4*)(fv + p * 4);
        sx += f.x; sy += f.y; sz += f.z;
      }
      *(float4*)&lds[wv][p * 4] = f;
    }
  }
  asm volatile("s_wait_dscnt 0" ::: "memory");     // wave-local LDS RAW fence

#pragma unroll
  for (int off = 16; off > 0; off >>= 1) {
    sx += __shfl_xor(sx, off, 32);
    sy += __shfl_xor(sy, off, 32);
    sz += __shfl_xor(sz, off, 32);
  }
  const float inv_n = 1.f / (float)npts;
  const float mx = sx * inv_n, my = sy * inv_n, mz = sz * inv_n;

  // --- per-pillar offset row, carried as K=4 of the GEMM ---------------------
  v2f b1f[4];
#pragma unroll
  for (int nt = 0; nt < 4; ++nt) {
    float off = -fmaf(cy, o8[nt], fmaf(cx, o7[nt],
                 fmaf(mz, o6[nt], fmaf(my, o5[nt], mx * o4[nt]))));
    b1f[nt][0] = hi ? 0.f : off;                   // B row K4 = off[n]; K5..7 = 0
    b1f[nt][1] = 0.f;
  }

  // LDS byte base for this lane's A-fragment; tiles index via immediate offset
  const char* lp = (const char*)&lds[0][0] +
                   ((wv * MPAD + nn) * 4 + (hi ? 2 : 0)) * sizeof(float);

  float runmax[4] = {NEG_INF, NEG_INF, NEG_INF, NEG_INF};

#pragma unroll
  for (int t = 0; t < 7; ++t) {
    // A kstep0: raw masked feats (lo lanes K0,1; hi lanes K2,3), ds OFFSET=t*256
    v2f a0 = *(const v2f*)(lp + t * 256);
    // A kstep1: real-row indicator in K4, zeros elsewhere
    v2f a1;
    a1[0] = (!hi && (t * 16 + nn) < npts) ? 1.f : 0.f;
    a1[1] = 0.f;

    v8f acc[4];
#pragma unroll
    for (int nt = 0; nt < 4; ++nt)
      acc[nt] = __builtin_amdgcn_wmma_f32_16x16x4_f32(
          false, a0, false, Bf0[nt], (short)0, v8f{}, false, false);
#pragma unroll
    for (int nt = 0; nt < 4; ++nt)
      acc[nt] = __builtin_amdgcn_wmma_f32_16x16x4_f32(
          false, a1, false, b1f[nt], (short)0, acc[nt], false, false);

    if (t < 6) {
      // all 16 rows are m<96<100: padded rows contribute 0 (correct), pure max
#pragma unroll
      for (int nt = 0; nt < 4; ++nt) {
        float m01 = fmaxf(acc[nt][0], acc[nt][1]);
        float m23 = fmaxf(acc[nt][2], acc[nt][3]);
        float m45 = fmaxf(acc[nt][4], acc[nt][5]);
        float m67 = fmaxf(acc[nt][6], acc[nt][7]);
        float mq  = fmaxf(fmaxf(m01, m23), fmaxf(m45, m67));
        runmax[nt] = fmaxf(runmax[nt], mq);
      }
    } else {
      // tile 6: valid rows are 96..99 -> lo lanes, r<4 only
#pragma unroll
      for (int nt = 0; nt < 4; ++nt) {
#pragma unroll
        for (int r = 0; r < 4; ++r) {
          float val = hi ? NEG_INF : acc[nt][r];
          runmax[nt] = fmaxf(runmax[nt], val);
        }
      }
    }
  }

  // combine lane-halves (M 0..7 vs 8..15), then bias + ReLU once per channel
#pragma unroll
  for (int nt = 0; nt < 4; ++nt) {
    runmax[nt] = fmaxf(runmax[nt], __shfl_xor(runmax[nt], 16, 32));
    runmax[nt] = fmaxf(runmax[nt] + biasn[nt], 0.f);
  }

  if (lane < 16) {
    float* o = out + (size_t)v * COUT + nn;
#pragma unroll
    for (int nt = 0; nt < 4; ++nt) o[nt * 16] = runmax[nt];
  }
}

// ------------------------------- launcher -----------------------------------
extern "C" void kernel_launch(void* const* d_in, const int* in_sizes, int n_in,
                              void* d_out, int out_size, void* d_ws, size_t ws_size,
                              hipStream_t stream) {
  const float* features   = (const float*)d_in[0];
  const int*   indices    = (const int*)  d_in[1];
  const int*   num_points = (const int*)  d_in[2];
  const float* W          = (const float*)d_in[3];
  const float* gamma      = (const float*)d_in[4];
  const float* beta       = (const float*)d_in[5];
  const float* rmean      = (const float*)d_in[6];
  const float* rvar       = (const float*)d_in[7];
  float*       ws         = (float*)d_ws;          // needs 768*4 = 3072 bytes
  float*       out        = (float*)d_out;

  pfn_prep<<<1, 256, 0, stream>>>(W, gamma, beta, rmean, rvar, ws);
  pfn_main<<<(NVOX + 7) / 8, 256, 0, stream>>>(features, indices, num_points, ws, out);
}